// MedianGCN_4930622456135
// MI455X (gfx1250) — compile-verified
//
#include <hip/hip_runtime.h>

typedef float v2f __attribute__((ext_vector_type(2)));
typedef float v8f __attribute__((ext_vector_type(8)));

#define N_NODES 30000
#define K_NBR   32
#define NFEAT   512
#define NHID    128
#define NCLASS  64

// ---------------------------------------------------------------------------
// Dense GEMM C[M,FOUT] = A[M,FIN] @ B[FIN,FOUT] via V_WMMA_F32_16X16X4_F32.
//
// One wave32 owns a 16x16 C tile; a block of FOUT/16 waves covers 16 full
// rows of C. The 16-row A slab (16 x KSLAB fp32) is staged cooperatively in
// LDS once per block (coalesced global_load_b128 fills), eliminating the
// 8x inter-wave redundancy of reading A straight from global. LDS row
// stride 68 dwords => ds_load_b64 fragment reads are bank-conflict-free
// (banks (4*row + k) % 64 are distinct across the 16 rows) and rows stay
// 16B-aligned for the b128 fills. Next slab is prefetched
// (global_prefetch_b8) while the current slab's 16 WMMAs issue.
//
// ISA wave32 layouts (CDNA5 ISA):
//   A 16x4  : lane L holds row m=L%16, k = {2*(L/16), 2*(L/16)+1}
//   B 4x16  : lane L holds col n=L%16, k = {2*(L/16), 2*(L/16)+1}
//   C 16x16 : VGPR r -> row r (lanes 0-15) / row r+8 (lanes 16-31), col L%16
// ---------------------------------------------------------------------------
template <int FIN, int FOUT>
__global__ __launch_bounds__(32 * (FOUT / 16)) void gemm_wmma_f32(
    const float* __restrict__ A, const float* __restrict__ B,
    float* __restrict__ C) {
  constexpr int NW      = FOUT / 16;  // waves per block
  constexpr int KSLAB   = 64;         // k-depth staged per iteration
  constexpr int ASTRIDE = 68;         // LDS row stride in dwords (pad)

  __shared__ float As[16 * ASTRIDE];

  const int t    = threadIdx.x;
  const int lane = t & 31;
  const int wave = t >> 5;
  const int m0   = blockIdx.x * 16;
  const int n0   = wave * 16;
  const int r    = lane & 15;         // A row / B,C col within tile
  const int kh   = (lane >> 4) << 1;  // 0 or 2: k-pair owned by this half

  v8f acc = {};
  const float* bcol = B + (size_t)kh * FOUT + (n0 + r);  // stride FOUT in k

  for (int k0 = 0; k0 < FIN; k0 += KSLAB) {
    __syncthreads();  // previous slab fully consumed before overwrite
    // Cooperative slab fill: 16 rows x (KSLAB/4) float4 quads = 256 tasks.
#pragma unroll
    for (int task = t; task < 16 * (KSLAB / 4); task += NW * 32) {
      const int row = task >> 4;   // KSLAB/4 == 16 quads per row
      const int q   = task & 15;
      const float4 src =
          *(const float4*)(A + (size_t)(m0 + row) * FIN + k0 + q * 4);
      *(float4*)(&As[row * ASTRIDE + q * 4]) = src;
    }
    __syncthreads();

    if (k0 + KSLAB < FIN)  // overlap next slab fetch with this slab's WMMAs
      __builtin_prefetch(A + (size_t)(m0 + r) * FIN + k0 + KSLAB, 0, 0);

    const float* as = &As[r * ASTRIDE + kh];
#pragma unroll
    for (int kk = 0; kk < KSLAB; kk += 4) {
      v2f a, b;
      a.x = as[kk];
      a.y = as[kk + 1];
      b.x = bcol[0];
      b.y = bcol[FOUT];
      // 8 args: (neg_a, A, neg_b, B, c_mod, C, reuse_a, reuse_b)
      acc = __builtin_amdgcn_wmma_f32_16x16x4_f32(
          false, a, false, b, (short)0, acc, false, false);
      bcol += 4 * FOUT;
    }
  }

  const int orow = m0 + ((lane >> 4) << 3);  // +0 or +8
  float* crow = C + (size_t)orow * FOUT + (n0 + r);
#pragma unroll
  for (int i = 0; i < 8; ++i) crow[(size_t)i * FOUT] = acc[i];
}

// ---------------------------------------------------------------------------
// Lower median of 33 gathered rows + bias (+ optional ReLU).
// One block per node, F threads (one per feature) -> each neighbor row
// h[src][0..F-1] is read fully coalesced by the block; neighbor indices are
// block-uniform (scalarizable). The 33-element array is kept entirely in
// VGPRs via a fully-unrolled 17-pass bubble selection (constant indices,
// v_min/v_max pairs are independent -> VOPD dual-issue), leaving
// v[16] = the lower median (sorted index (33-1)/2).
// ---------------------------------------------------------------------------
template <int F, bool RELU>
__global__ __launch_bounds__(F) void median_bias(
    const float* __restrict__ h, const int* __restrict__ nbr,
    const float* __restrict__ bias, float* __restrict__ out) {
  const int n = blockIdx.x;
  const int f = threadIdx.x;
  const int* nb = nbr + (size_t)n * K_NBR;

  float v[K_NBR + 1];
#pragma unroll 4
  for (int k = 0; k < K_NBR; ++k) {
    int src = n + nb[k] + 1;             // nb in [0, N-2] -> src in [1, 2N-2]
    if (src >= N_NODES) src -= N_NODES;  // single wrap suffices
    v[k] = h[(size_t)src * F + f];
  }
  v[K_NBR] = h[(size_t)n * F + f];       // self loop

#pragma unroll
  for (int i = 0; i <= K_NBR / 2; ++i) { // 17 passes: v[0..16] sorted prefix
#pragma unroll
    for (int j = K_NBR; j > i; --j) {
      float a = v[j - 1], b = v[j];
      v[j - 1] = fminf(a, b);
      v[j]     = fmaxf(a, b);
    }
  }

  float res = v[K_NBR / 2] + bias[f];
  out[(size_t)n * F + f] = RELU ? fmaxf(res, 0.0f) : res;
}

// ---------------------------------------------------------------------------
// Layer-2 median + bias fused with log_softmax over NCLASS=64.
// One block (64 threads) per node; shared-memory tree reductions for the
// max and sum(exp) of the 64 class logits.
// ---------------------------------------------------------------------------
__global__ __launch_bounds__(NCLASS) void median2_logsoftmax(
    const float* __restrict__ z, const int* __restrict__ nbr,
    const float* __restrict__ bias, float* __restrict__ out) {
  __shared__ float red[NCLASS];
  const int n = blockIdx.x;
  const int c = threadIdx.x;
  const int* nb = nbr + (size_t)n * K_NBR;

  float v[K_NBR + 1];
#pragma unroll 4
  for (int k = 0; k < K_NBR; ++k) {
    int src = n + nb[k] + 1;
    if (src >= N_NODES) src -= N_NODES;
    v[k] = z[(size_t)src * NCLASS + c];
  }
  v[K_NBR] = z[(size_t)n * NCLASS + c];

#pragma unroll
  for (int i = 0; i <= K_NBR / 2; ++i) {
#pragma unroll
    for (int j = K_NBR; j > i; --j) {
      float a = v[j - 1], b = v[j];
      v[j - 1] = fminf(a, b);
      v[j]     = fmaxf(a, b);
    }
  }
  const float med = v[K_NBR / 2] + bias[c];

  // max over classes
  red[c] = med;
  __syncthreads();
  for (int s = NCLASS / 2; s > 0; s >>= 1) {
    if (c < s) red[c] = fmaxf(red[c], red[c + s]);
    __syncthreads();
  }
  const float mx = red[0];
  __syncthreads();

  // sum of exp
  red[c] = expf(med - mx);
  __syncthreads();
  for (int s = NCLASS / 2; s > 0; s >>= 1) {
    if (c < s) red[c] += red[c + s];
    __syncthreads();
  }
  const float lse = mx + logf(red[0]);

  out[(size_t)n * NCLASS + c] = med - lse;
}

// ---------------------------------------------------------------------------
extern "C" void kernel_launch(void* const* d_in, const int* in_sizes, int n_in,
                              void* d_out, int out_size, void* d_ws,
                              size_t ws_size, hipStream_t stream) {
  const float* x   = (const float*)d_in[0];  // [N, NFEAT]
  const int*   nbr = (const int*)d_in[1];    // [N, K]
  const float* W1  = (const float*)d_in[2];  // [NFEAT, NHID]
  const float* b1  = (const float*)d_in[3];  // [NHID]
  const float* W2  = (const float*)d_in[4];  // [NHID, NCLASS]
  const float* b2  = (const float*)d_in[5];  // [NCLASS]
  float* out = (float*)d_out;                // [N, NCLASS]

  float* h1  = (float*)d_ws;                 // N*NHID raw GEMM1 output
  float* h1m = h1 + (size_t)N_NODES * NHID;  // N*NHID median+relu
  float* z2  = h1;                           // alias: h1 dead after median1

  gemm_wmma_f32<NFEAT, NHID>
      <<<N_NODES / 16, 32 * (NHID / 16), 0, stream>>>(x, W1, h1);

  median_bias<NHID, true>
      <<<N_NODES, NHID, 0, stream>>>(h1, nbr, b1, h1m);

  gemm_wmma_f32<NHID, NCLASS>
      <<<N_NODES / 16, 32 * (NCLASS / 16), 0, stream>>>(h1m, W2, z2);

  median2_logsoftmax<<<N_NODES, NCLASS, 0, stream>>>(z2, nbr, b2, out);
}